// KNNPoseDecoder_with_intrinsics_55070070670322
// MI455X (gfx1250) — compile-verified
//
#include <hip/hip_runtime.h>
#include <hip/hip_bf16.h>

typedef __attribute__((ext_vector_type(2))) float v2f;
typedef __attribute__((ext_vector_type(8))) float v8f;

#define LOG_HI_C 4.6051701859880914f
#define LOG_LO_C (-4.6051701859880914f)

// ---------------------------------------------------------------------------
// Squeeze 1x1 conv + ReLU.  GEMM M=256(oc) N=122880(b*480) K=512.
// 64(M) x 32(N) register-blocked tile per wave: 8 WMMAs per K=4 step.
// ---------------------------------------------------------------------------
__global__ __launch_bounds__(256) void squeeze_kernel(
    const float* __restrict__ f0, const float* __restrict__ W,
    const float* __restrict__ bias, float* __restrict__ cat)
{
  const int wave = threadIdx.x >> 5;
  const int lane = threadIdx.x & 31;
  const int ln   = lane & 15;
  const int half = lane >> 4;
  int tile = blockIdx.x * 8 + wave;           // 4 * 3840 = 15360 wave-tiles
  int tm = tile & 3;                          // 4 M supertiles (64 rows)
  int tn = tile >> 2;                         // 3840 N supertiles (32 cols)
  int b   = tn / 15;
  int hw0 = (tn % 15) * 32;
  int m0  = tm * 64;

  const float* Wrow = W + (size_t)(m0 + ln) * 512;        // + j*16*512
  const float* Xc0  = f0 + (size_t)b * 512 * 480 + hw0 + ln;
  const float* Xc1  = Xc0 + 16;

  v8f acc[4][2];
#pragma unroll
  for (int j = 0; j < 4; ++j) { acc[j][0] = (v8f){}; acc[j][1] = (v8f){}; }

  for (int k = 0; k < 512; k += 4) {
    int ka = k + 2 * half;
    v2f a[4], b0, b1;
#pragma unroll
    for (int j = 0; j < 4; ++j) {
      a[j].x = Wrow[(size_t)j * 16 * 512 + ka];
      a[j].y = Wrow[(size_t)j * 16 * 512 + ka + 1];
    }
    b0.x = Xc0[(size_t)ka * 480]; b0.y = Xc0[(size_t)(ka + 1) * 480];
    b1.x = Xc1[(size_t)ka * 480]; b1.y = Xc1[(size_t)(ka + 1) * 480];
#pragma unroll
    for (int j = 0; j < 4; ++j) {
      acc[j][0] = __builtin_amdgcn_wmma_f32_16x16x4_f32(false, a[j], false, b0,
                                                        (short)0, acc[j][0], false, false);
      acc[j][1] = __builtin_amdgcn_wmma_f32_16x16x4_f32(false, a[j], false, b1,
                                                        (short)0, acc[j][1], false, false);
    }
  }
  float* ob = cat + (size_t)b * 256 * 480 + hw0 + ln;
#pragma unroll
  for (int j = 0; j < 4; ++j)
#pragma unroll
    for (int r = 0; r < 8; ++r) {
      int m = m0 + j * 16 + r + 8 * half;
      float v0 = acc[j][0][r] + bias[m];
      float v1 = acc[j][1][r] + bias[m];
      ob[(size_t)m * 480]      = v0 > 0.f ? v0 : 0.f;
      ob[(size_t)m * 480 + 16] = v1 > 0.f ? v1 : 0.f;
    }
}

// Transpose conv weights [OC,IC,3,3] -> [tap][OC][IC] so K(=ic) is contiguous.
__global__ __launch_bounds__(256) void wtrans_kernel(
    const float* __restrict__ w, float* __restrict__ out, int IC, int OC)
{
  int idx = blockIdx.x * 256 + threadIdx.x;
  int total = OC * IC * 9;
  if (idx < total) {
    int oc  = idx / (IC * 9);
    int rem = idx % (IC * 9);
    int ic  = rem / 9;
    int tap = rem % 9;
    out[((size_t)tap * OC + oc) * IC + ic] = w[idx];
  }
}

// ---------------------------------------------------------------------------
// 3x3 conv (zero or reflect pad) + ReLU/ELU, implicit GEMM via f32 WMMA.
// Weights pre-transposed to [tap][OC][IC].  64(M) x 32(N) tile per wave.
// ---------------------------------------------------------------------------
__global__ __launch_bounds__(256) void conv3x3_wmma_kernel(
    const float* __restrict__ in, const float* __restrict__ Wt,
    const float* __restrict__ bias, float* __restrict__ out,
    int IC, int OC, int act /*0=relu,1=elu*/, int refl /*0=zero,1=reflect*/)
{
  const int wave = threadIdx.x >> 5;
  const int lane = threadIdx.x & 31;
  const int ln   = lane & 15;
  const int half = lane >> 4;
  int tile   = blockIdx.x * 8 + wave;
  int tilesM = OC >> 6;                       // 64-row supertiles
  int tm = tile % tilesM;
  int tn = tile / tilesM;                     // 3840 N supertiles
  int b   = tn / 15;
  int hw0 = (tn % 15) * 32;
  int m0  = tm * 64;

  int nA = hw0 + ln,      nB = hw0 + 16 + ln;
  int h0 = nA / 40, w0 = nA % 40;
  int h1 = nB / 40, w1 = nB % 40;

  const float* inb = in + (size_t)b * IC * 480;

  v8f acc[4][2];
#pragma unroll
  for (int j = 0; j < 4; ++j) { acc[j][0] = (v8f){}; acc[j][1] = (v8f){}; }

  for (int tap = 0; tap < 9; ++tap) {
    int dy = tap / 3 - 1, dx = tap % 3 - 1;
    int hiA = h0 + dy, wiA = w0 + dx;
    int hiB = h1 + dy, wiB = w1 + dx;
    float mA = 1.f, mB = 1.f;
    if (refl) {
      hiA = hiA < 0 ? -hiA : (hiA > 11 ? 22 - hiA : hiA);
      wiA = wiA < 0 ? -wiA : (wiA > 39 ? 78 - wiA : wiA);
      hiB = hiB < 0 ? -hiB : (hiB > 11 ? 22 - hiB : hiB);
      wiB = wiB < 0 ? -wiB : (wiB > 39 ? 78 - wiB : wiB);
    } else {
      bool vA = (hiA >= 0) && (hiA < 12) && (wiA >= 0) && (wiA < 40);
      bool vB = (hiB >= 0) && (hiB < 12) && (wiB >= 0) && (wiB < 40);
      mA = vA ? 1.f : 0.f; if (!vA) { hiA = 0; wiA = 0; }
      mB = vB ? 1.f : 0.f; if (!vB) { hiB = 0; wiB = 0; }
    }
    const float* xc0 = inb + hiA * 40 + wiA;
    const float* xc1 = inb + hiB * 40 + wiB;
    const float* wr  = Wt + ((size_t)tap * OC + m0 + ln) * IC;   // + j*16*IC

    for (int ic = 0; ic < IC; ic += 4) {
      int ka = ic + 2 * half;
      v2f a[4], b0, b1;
#pragma unroll
      for (int j = 0; j < 4; ++j)
        a[j] = *(const v2f*)(wr + (size_t)j * 16 * IC + ka);      // contiguous
      b0.x = xc0[(size_t)ka * 480] * mA; b0.y = xc0[(size_t)(ka + 1) * 480] * mA;
      b1.x = xc1[(size_t)ka * 480] * mB; b1.y = xc1[(size_t)(ka + 1) * 480] * mB;
#pragma unroll
      for (int j = 0; j < 4; ++j) {
        acc[j][0] = __builtin_amdgcn_wmma_f32_16x16x4_f32(false, a[j], false, b0,
                                                          (short)0, acc[j][0], false, false);
        acc[j][1] = __builtin_amdgcn_wmma_f32_16x16x4_f32(false, a[j], false, b1,
                                                          (short)0, acc[j][1], false, false);
      }
    }
  }
  float* ob = out + (size_t)b * OC * 480 + hw0 + ln;
#pragma unroll
  for (int j = 0; j < 4; ++j)
#pragma unroll
    for (int r = 0; r < 8; ++r) {
      int m = m0 + j * 16 + r + 8 * half;
      float v0 = acc[j][0][r] + bias[m];
      float v1 = acc[j][1][r] + bias[m];
      if (act == 0) { v0 = fmaxf(v0, 0.f); v1 = fmaxf(v1, 0.f); }
      else { v0 = v0 > 0.f ? v0 : expm1f(v0); v1 = v1 > 0.f ? v1 : expm1f(v1); }
      ob[(size_t)m * 480]      = v0;
      ob[(size_t)m * 480 + 16] = v1;
    }
}

// ---------------------------------------------------------------------------
// FC via WMMA: out[m,n] = act(sum_k A[m,k]*Wt[n,k] + b[n]); M=N=256.
// 64x32 tile per wave -> 32 waves total.
// ---------------------------------------------------------------------------
__global__ __launch_bounds__(256) void fc_wmma_kernel(
    const float* __restrict__ A, const float* __restrict__ Wt,
    const float* __restrict__ bias, float* __restrict__ out, int K, int relu)
{
  const int wave = threadIdx.x >> 5;
  const int lane = threadIdx.x & 31;
  const int ln   = lane & 15;
  const int half = lane >> 4;
  int tile = blockIdx.x * 8 + wave;           // 32 tiles
  int tm = tile >> 3, tn = tile & 7;
  int m0 = tm * 64, n0 = tn * 32;
  const float* Ar  = A  + (size_t)(m0 + ln) * K;
  const float* Br0 = Wt + (size_t)(n0 + ln) * K;
  const float* Br1 = Br0 + (size_t)16 * K;

  v8f acc[4][2];
#pragma unroll
  for (int j = 0; j < 4; ++j) { acc[j][0] = (v8f){}; acc[j][1] = (v8f){}; }

  for (int k = 0; k < K; k += 4) {
    int kk = k + 2 * half;
    v2f a[4], b0, b1;
#pragma unroll
    for (int j = 0; j < 4; ++j)
      a[j] = *(const v2f*)(Ar + (size_t)j * 16 * K + kk);
    b0 = *(const v2f*)(Br0 + kk);
    b1 = *(const v2f*)(Br1 + kk);
#pragma unroll
    for (int j = 0; j < 4; ++j) {
      acc[j][0] = __builtin_amdgcn_wmma_f32_16x16x4_f32(false, a[j], false, b0,
                                                        (short)0, acc[j][0], false, false);
      acc[j][1] = __builtin_amdgcn_wmma_f32_16x16x4_f32(false, a[j], false, b1,
                                                        (short)0, acc[j][1], false, false);
    }
  }
#pragma unroll
  for (int j = 0; j < 4; ++j)
#pragma unroll
    for (int r = 0; r < 8; ++r) {
      int m = m0 + j * 16 + r + 8 * half;
      int n = n0 + ln;
      float v0 = acc[j][0][r] + bias[n];
      float v1 = acc[j][1][r] + bias[n + 16];
      if (relu) { v0 = fmaxf(v0, 0.f); v1 = fmaxf(v1, 0.f); }
      out[(size_t)m * 256 + n]      = v0;
      out[(size_t)m * 256 + n + 16] = v1;
    }
}

// ---------------------------------------------------------------------------
// Row mean over last dim L: in[rows][L] -> out[rows]
// ---------------------------------------------------------------------------
__global__ __launch_bounds__(256) void rowmean_kernel(
    const float* __restrict__ in, float* __restrict__ out, int L)
{
  __shared__ float s[256];
  int row = blockIdx.x;
  const float* p = in + (size_t)row * L;
  float sum = 0.f;
  for (int i = threadIdx.x; i < L; i += 256) sum += p[i];
  s[threadIdx.x] = sum;
  __syncthreads();
  for (int off = 128; off > 0; off >>= 1) {
    if (threadIdx.x < off) s[threadIdx.x] += s[threadIdx.x + off];
    __syncthreads();
  }
  if (threadIdx.x == 0) out[row] = s[0] / (float)L;
}

// Bank row L2 norms (1000 x 256)
__global__ __launch_bounds__(256) void banknorm_kernel(
    const float* __restrict__ bank, float* __restrict__ norms)
{
  __shared__ float s[256];
  int row = blockIdx.x;
  float v = bank[(size_t)row * 256 + threadIdx.x];
  s[threadIdx.x] = v * v;
  __syncthreads();
  for (int off = 128; off > 0; off >>= 1) {
    if (threadIdx.x < off) s[threadIdx.x] += s[threadIdx.x + off];
    __syncthreads();
  }
  if (threadIdx.x == 0) norms[row] = fmaxf(sqrtf(s[0]), 1e-12f);
}

// ---------------------------------------------------------------------------
// KNN: cosine sims vs bank, top-5, softmax, weighted neighbor sum.
// ---------------------------------------------------------------------------
__global__ __launch_bounds__(256) void knn_kernel(
    const float* __restrict__ pooled, const float* __restrict__ bank,
    const float* __restrict__ bnorm, float* __restrict__ fused)
{
  __shared__ float sp[256];
  __shared__ float sims[1000];
  __shared__ float red[256];
  __shared__ int   redi[256];
  __shared__ int   topi[5];
  __shared__ float topv[5];
  __shared__ float w5[5];
  int b = blockIdx.x, t = threadIdx.x;
  sp[t] = pooled[(size_t)b * 256 + t];
  __syncthreads();
  red[t] = sp[t] * sp[t];
  __syncthreads();
  for (int off = 128; off > 0; off >>= 1) {
    if (t < off) red[t] += red[t + off];
    __syncthreads();
  }
  float qn = fmaxf(sqrtf(red[0]), 1e-12f);
  __syncthreads();
  for (int j = t; j < 1000; j += 256) {
    const float* br = bank + (size_t)j * 256;
    float d = 0.f;
    for (int k = 0; k < 256; ++k) d += sp[k] * br[k];
    sims[j] = d / (qn * bnorm[j]);
  }
  __syncthreads();
  for (int it = 0; it < 5; ++it) {
    float best = -1e38f; int bi = 0;
    for (int j = t; j < 1000; j += 256) {
      float v = sims[j];
      if (v > best) { best = v; bi = j; }
    }
    red[t] = best; redi[t] = bi;
    __syncthreads();
    for (int off = 128; off > 0; off >>= 1) {
      if (t < off && red[t + off] > red[t]) { red[t] = red[t + off]; redi[t] = redi[t + off]; }
      __syncthreads();
    }
    if (t == 0) { topv[it] = red[0]; topi[it] = redi[0]; sims[redi[0]] = -1e30f; }
    __syncthreads();
  }
  if (t == 0) {
    float mx = topv[0], sum = 0.f;
    for (int k = 0; k < 5; ++k) { w5[k] = expf(topv[k] - mx); sum += w5[k]; }
    for (int k = 0; k < 5; ++k) w5[k] /= sum;
  }
  __syncthreads();
  float wn = 0.f;
  for (int k = 0; k < 5; ++k) wn += w5[k] * bank[(size_t)topi[k] * 256 + t];
  fused[(size_t)b * 512 + t]       = sp[t];
  fused[(size_t)b * 512 + 256 + t] = wn;
}

// cat[b,c,hw] += 0.1 * proc[b,c]
__global__ __launch_bounds__(256) void catupdate_kernel(
    float* __restrict__ cat, const float* __restrict__ proc)
{
  size_t i = (size_t)blockIdx.x * 256 + threadIdx.x;
  size_t bc = i / 480;
  cat[i] += 0.1f * proc[bc];
}

// out12[b,oc] = sum_ic W[oc,ic]*m2[b,ic] + bias[oc]
__global__ void pose2_kernel(const float* __restrict__ m2, const float* __restrict__ W,
                             const float* __restrict__ bias, float* __restrict__ out12)
{
  int b = blockIdx.x, oc = threadIdx.x;
  if (oc < 12) {
    const float* mm = m2 + (size_t)b * 256;
    const float* wr = W + (size_t)oc * 256;
    float s = 0.f;
    for (int k = 0; k < 256; ++k) s += mm[k] * wr[k];
    out12[b * 12 + oc] = s + bias[oc];
  }
}

__global__ void scalehead_kernel(const float* __restrict__ smean,
                                 const float* __restrict__ lw, const float* __restrict__ lb,
                                 float* __restrict__ scaleBuf, float* __restrict__ outScale)
{
  int b = blockIdx.x * blockDim.x + threadIdx.x;
  if (b < 256) {
    float s = 0.f;
    const float* sm = smean + (size_t)b * 64;
    for (int k = 0; k < 64; ++k) s += sm[k] * lw[k];
    s += lb[0];
    float sig = 1.f / (1.f + expf(-s));
    float sc = expf(sig * (LOG_HI_C - LOG_LO_C) + LOG_LO_C);
    scaleBuf[b] = sc;
    outScale[b] = sc;
  }
}

__global__ void final_kernel(const float* __restrict__ out12,
                             const float* __restrict__ scaleBuf, float* __restrict__ dout)
{
  int i = blockIdx.x * blockDim.x + threadIdx.x;
  if (i < 1536) {
    int b = i / 6, rem = i % 6, r = rem / 3, j = rem % 3;
    float sc = scaleBuf[b];
    dout[i]        = 0.001f * out12[b * 12 + r * 6 + j]     * sc;
    dout[1536 + i] = 0.001f * out12[b * 12 + r * 6 + 3 + j] * sc;
  }
}

extern "C" void kernel_launch(void* const* d_in, const int* in_sizes, int n_in,
                              void* d_out, int out_size, void* d_ws, size_t ws_size,
                              hipStream_t stream)
{
  const float* f0        = (const float*)d_in[0];
  const float* squeeze_w = (const float*)d_in[1];
  const float* squeeze_b = (const float*)d_in[2];
  const float* pose0_w   = (const float*)d_in[3];
  const float* pose0_b   = (const float*)d_in[4];
  const float* pose1_w   = (const float*)d_in[5];
  const float* pose1_b   = (const float*)d_in[6];
  const float* pose2_w   = (const float*)d_in[7];
  const float* pose2_b   = (const float*)d_in[8];
  const float* fusion_w1 = (const float*)d_in[9];
  const float* fusion_b1 = (const float*)d_in[10];
  const float* fusion_w2 = (const float*)d_in[11];
  const float* fusion_b2 = (const float*)d_in[12];
  const float* bank      = (const float*)d_in[13];
  const float* scale_w1  = (const float*)d_in[14];
  const float* scale_b1  = (const float*)d_in[15];
  const float* scale_w2  = (const float*)d_in[16];
  const float* scale_b2  = (const float*)d_in[17];
  const float* scale_lw  = (const float*)d_in[18];
  const float* scale_lb  = (const float*)d_in[19];
  float* out = (float*)d_out;

  const size_t BIG = (size_t)256 * 256 * 480;          // 31,457,280 floats
  float* ws    = (float*)d_ws;
  float* cat   = ws;
  float* tmp1  = ws + BIG;
  float* tmp2  = ws + 2 * BIG;
  float* small = ws + 3 * BIG;
  float* pooled = small;            small += 65536;    // 256*256
  float* bnorm  = small;            small += 1024;     // 1000
  float* fused  = small;            small += 131072;   // 256*512
  float* hbuf   = small;            small += 65536;
  float* proc   = small;            small += 65536;
  float* m2     = small;            small += 65536;
  float* out12  = small;            small += 3072;
  float* smean  = small;            small += 16384;
  float* scaleB = small;            small += 256;
  float* wT     = small;            small += 589824;   // transposed conv weights (max 256*256*9)

  // 1) squeeze 1x1 conv + relu -> cat   (4*3840 wave-tiles / 8 per block)
  squeeze_kernel<<<1920, 256, 0, stream>>>(f0, squeeze_w, squeeze_b, cat);

  // 2) pooled = spatial mean of cat
  rowmean_kernel<<<65536, 256, 0, stream>>>(cat, pooled, 480);

  // 3/4) bank norms, KNN -> fused
  banknorm_kernel<<<1000, 256, 0, stream>>>(bank, bnorm);
  knn_kernel<<<256, 256, 0, stream>>>(pooled, bank, bnorm, fused);

  // 5/6) fusion MLP (WMMA)
  fc_wmma_kernel<<<4, 256, 0, stream>>>(fused, fusion_w1, fusion_b1, hbuf, 512, 1);
  fc_wmma_kernel<<<4, 256, 0, stream>>>(hbuf, fusion_w2, fusion_b2, proc, 256, 0);

  // 7) cat += 0.1 * proc broadcast
  catupdate_kernel<<<122880, 256, 0, stream>>>(cat, proc);

  // 8/9) pose convs (zero pad, relu)
  wtrans_kernel<<<(256 * 256 * 9 + 255) / 256, 256, 0, stream>>>(pose0_w, wT, 256, 256);
  conv3x3_wmma_kernel<<<1920, 256, 0, stream>>>(cat,  wT, pose0_b, tmp1, 256, 256, 0, 0);
  wtrans_kernel<<<(256 * 256 * 9 + 255) / 256, 256, 0, stream>>>(pose1_w, wT, 256, 256);
  conv3x3_wmma_kernel<<<1920, 256, 0, stream>>>(tmp1, wT, pose1_b, tmp2, 256, 256, 0, 0);

  // 10/11) pose2 1x1-conv+mean collapsed to (channel mean) x W
  rowmean_kernel<<<65536, 256, 0, stream>>>(tmp2, m2, 480);
  pose2_kernel<<<256, 16, 0, stream>>>(m2, pose2_w, pose2_b, out12);

  // 12/13) scale convs (reflect pad, elu)
  wtrans_kernel<<<(128 * 256 * 9 + 255) / 256, 256, 0, stream>>>(scale_w1, wT, 256, 128);
  conv3x3_wmma_kernel<<<960, 256, 0, stream>>>(cat,  wT, scale_b1, tmp1, 256, 128, 1, 1);
  wtrans_kernel<<<(64 * 128 * 9 + 255) / 256, 256, 0, stream>>>(scale_w2, wT, 128, 64);
  conv3x3_wmma_kernel<<<480, 256, 0, stream>>>(tmp1, wT, scale_b2, tmp2, 128, 64, 1, 1);

  // 14) spatial mean of sfeat
  rowmean_kernel<<<16384, 256, 0, stream>>>(tmp2, smean, 480);

  // 15) scale head
  scalehead_kernel<<<1, 256, 0, stream>>>(smean, scale_lw, scale_lb, scaleB, out + 3072);

  // 16) axisangle / translation
  final_kernel<<<6, 256, 0, stream>>>(out12, scaleB, out);
}